// M_833223656318
// MI455X (gfx1250) — compile-verified
//
#include <hip/hip_runtime.h>
#include <cmath>

typedef __attribute__((ext_vector_type(2))) float v2f;
typedef __attribute__((ext_vector_type(4))) float v4f;
typedef __attribute__((ext_vector_type(8))) float v8f;

#define TM 128            // rows per block
#define LDS_STRIDE 388    // 384 + 4 floats pad -> conflict-free ds_load_b64 (16B aligned rows)
#define KDIM 384
#define ODIM 1536

// Branchless exact GELU: 0.5*y*(1+erf(y/sqrt2)), erf via Abramowitz-Stegun
// 7.1.26 (max abs err ~1.5e-7). One v_rcp + one v_exp + 8 fma, no EXEC
// divergence.
__device__ __forceinline__ float gelu_exact(float y) {
    const float x  = y * 0.70710678118654752440f;   // y / sqrt(2)
    const float ax = __builtin_fabsf(x);
    const float t  = __builtin_amdgcn_rcpf(__builtin_fmaf(0.3275911f, ax, 1.0f));
    float p = __builtin_fmaf(1.061405429f, t, -1.453152027f);
    p = __builtin_fmaf(p, t, 1.421413741f);
    p = __builtin_fmaf(p, t, -0.284496736f);
    p = __builtin_fmaf(p, t, 0.254829592f);
    p = p * t;
    const float e      = __expf(-x * x);            // v_exp_f32 (trans unit)
    const float erf_ax = __builtin_fmaf(-p, e, 1.0f);
    const float erf_x  = __builtin_copysignf(erf_ax, x);
    return 0.5f * y * (1.0f + erf_x);
}

__global__ __launch_bounds__(256) void ln_gemm_gelu_kernel(
    const float* __restrict__ x,
    const float* __restrict__ ln_w,
    const float* __restrict__ ln_b,
    const float* __restrict__ W,
    const float* __restrict__ bias,
    float* __restrict__ out)
{
    __shared__ float lds[TM * LDS_STRIDE];

    const int tid  = threadIdx.x;
    const int wave = tid >> 5;
    const int lane = tid & 31;
    const int row0 = blockIdx.x * TM;

    // ---------------- Phase 1: LayerNorm, 16 rows per wave (b128 vectorized) ----
    for (int r = 0; r < 16; ++r) {
        const int lrow = wave * 16 + r;
        const float* xr = x + (size_t)(row0 + lrow) * KDIM;
        v4f v[3];
        float s = 0.0f, ss = 0.0f;
#pragma unroll
        for (int i = 0; i < 3; ++i) {
            v[i] = *(const v4f*)(xr + lane * 4 + i * 128);   // global_load_b128, coalesced
#pragma unroll
            for (int j = 0; j < 4; ++j) {
                s  += v[i][j];
                ss += v[i][j] * v[i][j];
            }
        }
#pragma unroll
        for (int off = 16; off > 0; off >>= 1) {
            s  += __shfl_xor(s,  off, 32);
            ss += __shfl_xor(ss, off, 32);
        }
        const float mu  = s * (1.0f / KDIM);
        const float var = ss * (1.0f / KDIM) - mu * mu;
        const float rs  = rsqrtf(var + 1e-5f);
        float* dst = lds + lrow * LDS_STRIDE;
#pragma unroll
        for (int i = 0; i < 3; ++i) {
            const int k0 = lane * 4 + i * 128;
            const v4f lw = *(const v4f*)(ln_w + k0);
            const v4f lb = *(const v4f*)(ln_b + k0);
            v4f o;
#pragma unroll
            for (int j = 0; j < 4; ++j)
                o[j] = (v[i][j] - mu) * rs * lw[j] + lb[j];
            *(v4f*)(dst + k0) = o;                           // ds_store_b128, 16B aligned
        }
    }
    __syncthreads();

    // ---------------- Phase 2: WMMA f32 GEMM + bias + exact GELU ----------------
    // 8 waves = 2 (M halves of 64 rows) x 4 (col quarters of 384 outputs)
    const int mw   = wave & 1;
    const int nw   = wave >> 1;
    const int mrow = lane & 15;          // A fragment M index
    const int koff = (lane >> 4) << 1;   // A/B fragment K sub-offset {0,2}
    const int ncol = lane & 15;          // B/D fragment N index

    const float* ldsA = lds + (mw * 64 + mrow) * LDS_STRIDE + koff;

    for (int ct = 0; ct < 24; ++ct) {
        const int o0 = (nw * 24 + ct) << 4;
        const float bv = bias[o0 + ncol];                   // in flight during GEMM
        v8f acc0 = {}, acc1 = {}, acc2 = {}, acc3 = {};
        const float* wp = W + (size_t)(o0 + ncol) * KDIM + koff;
#pragma unroll 4
        for (int ks = 0; ks < 96; ++ks) {
            const int k4 = ks << 2;
            const v2f bf = *(const v2f*)(wp + k4);
            const v2f a0 = *(const v2f*)(ldsA + 0 * 16 * LDS_STRIDE + k4);
            const v2f a1 = *(const v2f*)(ldsA + 1 * 16 * LDS_STRIDE + k4);
            const v2f a2 = *(const v2f*)(ldsA + 2 * 16 * LDS_STRIDE + k4);
            const v2f a3 = *(const v2f*)(ldsA + 3 * 16 * LDS_STRIDE + k4);
            acc0 = __builtin_amdgcn_wmma_f32_16x16x4_f32(false, a0, false, bf, (short)0, acc0, false, false);
            acc1 = __builtin_amdgcn_wmma_f32_16x16x4_f32(false, a1, false, bf, (short)0, acc1, false, false);
            acc2 = __builtin_amdgcn_wmma_f32_16x16x4_f32(false, a2, false, bf, (short)0, acc2, false, false);
            acc3 = __builtin_amdgcn_wmma_f32_16x16x4_f32(false, a3, false, bf, (short)0, acc3, false, false);
        }

        const int rbase = row0 + mw * 64 + ((lane >> 4) << 3);

        auto store_tile = [&](v8f a, int s) {
            const size_t base = (size_t)(rbase + s * 16) * ODIM + (size_t)(o0 + ncol);
#pragma unroll
            for (int j = 0; j < 8; ++j) {
                out[base + (size_t)j * ODIM] = gelu_exact(a[j] + bv);
            }
        };
        store_tile(acc0, 0);
        store_tile(acc1, 1);
        store_tile(acc2, 2);
        store_tile(acc3, 3);
    }
}

extern "C" void kernel_launch(void* const* d_in, const int* in_sizes, int n_in,
                              void* d_out, int out_size, void* d_ws, size_t ws_size,
                              hipStream_t stream) {
    const float* x    = (const float*)d_in[0];
    const float* ln_w = (const float*)d_in[1];
    const float* ln_b = (const float*)d_in[2];
    const float* W    = (const float*)d_in[3];
    const float* b    = (const float*)d_in[4];
    float* out = (float*)d_out;

    const int rows = 256 * 14 * 14;   // 50176 = 392 * TM
    dim3 grid(rows / TM);
    dim3 block(256);
    ln_gemm_gelu_kernel<<<grid, block, 0, stream>>>(x, ln_w, ln_b, W, b, out);
}